// TripletSemihardLoss_48730698940868
// MI455X (gfx1250) — compile-verified
//
#include <hip/hip_runtime.h>

// Triplet semi-hard loss for N=384, D=128 on gfx1250 (MI455X).
// Gram matrix via V_WMMA_F32_16X16X4_F32 (full f32 precision so distance
// orderings match the reference); the [N,N,N] reduction is collapsed to
// per-row O(N) scans over the positive pairs only. All reductions are
// fixed-order (deterministic), no float atomics.

typedef __attribute__((ext_vector_type(2))) float v2f;
typedef __attribute__((ext_vector_type(8))) float v8f;

#define NN   384
#define DD   128
#define EPSF 1e-4f
#define MARGINF 1.0f

// ---------------------------------------------------------------------------
// Kernel 1: prod = E * E^T via f32 WMMA. One wave (32 threads) per 16x16 tile.
// A-layout (16x4 f32): lane<16 -> {K0,K1}, lane>=16 -> {K2,K3}, M = lane&15.
// B-layout (4x16 f32): symmetric (N = lane&15), and B[k][n] = E[n][k], so both
// operands load identically from E with different row bases.
// C/D layout: VGPR r, lanes 0-15 -> M=r; lanes 16-31 -> M=r+8; N = lane&15.
// ---------------------------------------------------------------------------
__global__ void gram_wmma_kernel(const float* __restrict__ E,
                                 float* __restrict__ prod) {
  const int lane = threadIdx.x;            // 0..31
  const int m    = lane & 15;
  const int kb   = (lane >> 4) << 1;       // 0 or 2
  const int ti   = blockIdx.x;
  const int tj   = blockIdx.y;

  const float* __restrict__ arow = E + (ti * 16 + m) * DD + kb;
  const float* __restrict__ brow = E + (tj * 16 + m) * DD + kb;

  v8f c = {0.f, 0.f, 0.f, 0.f, 0.f, 0.f, 0.f, 0.f};
#pragma unroll 8
  for (int k4 = 0; k4 < DD; k4 += 4) {
    v2f a = *(const v2f*)(arow + k4);
    v2f b = *(const v2f*)(brow + k4);
    // D = A*B + C   (v_wmma_f32_16x16x4_f32)
    c = __builtin_amdgcn_wmma_f32_16x16x4_f32(
        /*neg_a=*/false, a, /*neg_b=*/false, b,
        /*c_mod=*/(short)0, c, /*reuse_a=*/false, /*reuse_b=*/false);
  }

  const int row0 = ti * 16 + ((lane >> 4) << 3);
  const int col  = tj * 16 + m;
#pragma unroll
  for (int r = 0; r < 8; ++r)
    prod[(row0 + r) * NN + col] = c[r];
}

// ---------------------------------------------------------------------------
// Kernel 2: d[i,j] = sqrt(max(sq[i] + sq[j] - 2*prod[i,j], EPS))
// (clip(...,0) followed by clip(...,EPS) == max(...,EPS) since EPS > 0)
// ---------------------------------------------------------------------------
__global__ void dist_kernel(const float* __restrict__ prod,
                            float* __restrict__ dmat) {
  int idx = blockIdx.x * 256 + threadIdx.x;
  if (idx >= NN * NN) return;
  int i = idx / NN;
  int j = idx - i * NN;
  float sqi = prod[i * NN + i];
  float sqj = prod[j * NN + j];
  float p   = prod[idx];
  float r   = (sqi + sqj) - 2.0f * p;
  r = fmaxf(r, 0.0f);
  r = fmaxf(r, EPSF);
  dmat[idx] = sqrtf(r);
}

// ---------------------------------------------------------------------------
// Kernel 3: per-row semi-hard loss partials. One 384-thread block per row i.
// Thread t == column k. Only positive pairs (same label, k != i) contribute.
// Reproduces the reference's masked_maximum / masked_minimum arithmetic.
// ---------------------------------------------------------------------------
__global__ void row_loss_kernel(const float* __restrict__ dmat,
                                const int* __restrict__ labels,
                                float* __restrict__ pnum,
                                float* __restrict__ pcnt) {
  __shared__ float v[NN];
  __shared__ int   lab[NN];
  __shared__ float red[512];

  const int i = blockIdx.x;
  const int t = threadIdx.x;

  v[t]   = dmat[i * NN + t];
  lab[t] = labels[t];
  __syncthreads();

  const int   li = lab[i];
  const float vk = v[t];
  const bool  is_pos = (lab[t] == li) && (t != i);

  float num = 0.0f;
  float cnt = 0.0f;
  if (is_pos) {
    cnt = 1.0f;
    // axis_min / axis_max over j of dist_diff = vk - v[j]
    float axis_min =  3.4e38f;
    float axis_max = -3.4e38f;
    for (int j = 0; j < NN; ++j) {
      float diff = vk - v[j];
      axis_min = fminf(axis_min, diff);
      axis_max = fmaxf(axis_max, diff);
    }
    // masked reductions (masked-out entries contribute 0, exactly as in ref)
    float maxo = -3.4e38f;   // max_j (diff - axis_min) * mask
    float mino =  3.4e38f;   // min_j (diff - axis_max) * M
    bool  any  = false;      // mask_final = sum(mask) > 0
    for (int j = 0; j < NN; ++j) {
      float diff = vk - v[j];
      bool  Mj   = (lab[j] != li) || (j == i);     // 1 - pos[i,j]
      bool  mj   = Mj && (diff < 0.0f);
      any |= mj;
      float to = mj ? (diff - axis_min) : 0.0f;
      maxo = fmaxf(maxo, to);
      float tn = Mj ? (diff - axis_max) : 0.0f;
      mino = fminf(mino, tn);
    }
    float neg_outside = maxo + axis_min;
    float neg_inside  = mino + axis_max;
    float semi = any ? neg_outside : neg_inside;
    num = fmaxf(semi + MARGINF, 0.0f);             // relu(loss_mat), pos==1
  }

  // deterministic fixed-order block reduction (pad 384 -> 512)
  red[t] = num;
  if (t < 512 - NN) red[NN + t] = 0.0f;
  __syncthreads();
  for (int s = 256; s > 0; s >>= 1) {
    if (t < s) red[t] += red[t + s];
    __syncthreads();
  }
  if (t == 0) pnum[i] = red[0];
  __syncthreads();

  red[t] = cnt;
  if (t < 512 - NN) red[NN + t] = 0.0f;
  __syncthreads();
  for (int s = 256; s > 0; s >>= 1) {
    if (t < s) red[t] += red[t + s];
    __syncthreads();
  }
  if (t == 0) pcnt[i] = red[0];
}

// ---------------------------------------------------------------------------
// Kernel 4: final deterministic reduction of 384 row partials, write loss.
// ---------------------------------------------------------------------------
__global__ void finalize_kernel(const float* __restrict__ pnum,
                                const float* __restrict__ pcnt,
                                float* __restrict__ out) {
  __shared__ float red[512];
  const int t = threadIdx.x;

  red[t] = (t < NN) ? pnum[t] : 0.0f;
  __syncthreads();
  for (int s = 256; s > 0; s >>= 1) {
    if (t < s) red[t] += red[t + s];
    __syncthreads();
  }
  float num = red[0];
  __syncthreads();

  red[t] = (t < NN) ? pcnt[t] : 0.0f;
  __syncthreads();
  for (int s = 256; s > 0; s >>= 1) {
    if (t < s) red[t] += red[t + s];
    __syncthreads();
  }
  if (t == 0) out[0] = num / red[0];
}

// ---------------------------------------------------------------------------
extern "C" void kernel_launch(void* const* d_in, const int* in_sizes, int n_in,
                              void* d_out, int out_size, void* d_ws, size_t ws_size,
                              hipStream_t stream) {
  (void)in_sizes; (void)n_in; (void)out_size; (void)ws_size;

  const float* E      = (const float*)d_in[0];   // [384,128] f32
  const int*   labels = (const int*)d_in[1];     // [384] int
  float*       out    = (float*)d_out;           // scalar loss

  float* prod = (float*)d_ws;                    // [384*384]
  float* dmat = prod + NN * NN;                  // [384*384]
  float* pnum = dmat + NN * NN;                  // [384]
  float* pcnt = pnum + NN;                       // [384]

  gram_wmma_kernel<<<dim3(24, 24, 1), dim3(32, 1, 1), 0, stream>>>(E, prod);
  dist_kernel<<<(NN * NN + 255) / 256, 256, 0, stream>>>(prod, dmat);
  row_loss_kernel<<<NN, NN, 0, stream>>>(dmat, labels, pnum, pcnt);
  finalize_kernel<<<1, 512, 0, stream>>>(pnum, pcnt, out);
}